// gcn_encoder_29059748724761
// MI455X (gfx1250) — compile-verified
//
#include <hip/hip_runtime.h>
#include <hip/hip_bf16.h>

#define NN 5724
#define BSZ 2
#define EE (NN * 32)
#define DD 8
#define HH 64
#define LAYER 3
#define LN_EPS 1e-5f
#define NCHUNK ((NN + 15) / 16)   // 358 key chunks
#define SPLITS 4
#define CPS ((NCHUNK + SPLITS - 1) / SPLITS)  // 90 chunks per split

typedef __attribute__((ext_vector_type(2))) float v2f;
typedef __attribute__((ext_vector_type(8))) float v8f;

__device__ __forceinline__ v8f wmma4(v2f a, v2f b, v8f c) {
  // D(16x16,f32) = A(16x4,f32) * B(4x16,f32) + C
  return __builtin_amdgcn_wmma_f32_16x16x4_f32(false, a, false, b, (short)0, c,
                                               false, false);
}

__device__ __forceinline__ v8f vzero8() {
  v8f z = {0.f, 0.f, 0.f, 0.f, 0.f, 0.f, 0.f, 0.f};
  return z;
}

// ---------------------------------------------------------------------------
// Kernel 1: v2v = x*lin_w + lin_b ; q/k/v = v2v @ w{q,k,v}[3] + b (layer 3 only;
// layers 0..2 of the reference are dead code since each consumes v2v).
// ---------------------------------------------------------------------------
__global__ void qkv_kernel(const float* __restrict__ x,
                           const float* __restrict__ lin_w,
                           const float* __restrict__ lin_b,
                           const float* __restrict__ wq, const float* __restrict__ bq,
                           const float* __restrict__ wk, const float* __restrict__ bk,
                           const float* __restrict__ wv, const float* __restrict__ bv,
                           float* __restrict__ v2v, float* __restrict__ qo,
                           float* __restrict__ ko, float* __restrict__ vo) {
  int idx = blockIdx.x * blockDim.x + threadIdx.x;
  if (idx >= BSZ * NN) return;
  const float xs = x[idx];
  const float* WQ = wq + LAYER * 64; const float* BQ = bq + LAYER * 8;
  const float* WK = wk + LAYER * 64; const float* BK = bk + LAYER * 8;
  const float* WV = wv + LAYER * 64; const float* BV = bv + LAYER * 8;
  float h[8];
#pragma unroll
  for (int d = 0; d < 8; ++d) h[d] = xs * lin_w[d] + lin_b[d];
#pragma unroll
  for (int d = 0; d < 8; ++d) {
    float q = BQ[d], k = BK[d], v = BV[d];
#pragma unroll
    for (int t = 0; t < 8; ++t) {
      q += h[t] * WQ[t * 8 + d];
      k += h[t] * WK[t * 8 + d];
      v += h[t] * WV[t * 8 + d];
    }
    qo[(size_t)idx * 8 + d] = q;
    ko[(size_t)idx * 8 + d] = k;
    vo[(size_t)idx * 8 + d] = v;
    v2v[(size_t)idx * 8 + d] = h[d];
  }
}

// ---------------------------------------------------------------------------
// Kernel 2: split-K streaming attention via V_WMMA_F32_16X16X4_F32.
// grid = (query tiles of 32, SPLITS, BSZ); each wave covers a key range and
// emits unnormalized partials (O, m, l). S^T = K_tile*Q^T layout keeps the
// softmax per-query state in per-lane scalars.
// ---------------------------------------------------------------------------
__global__ __launch_bounds__(32) void attn_kernel(const float* __restrict__ Qg,
                                                  const float* __restrict__ Kg,
                                                  const float* __restrict__ Vg,
                                                  float* __restrict__ Part) {
  const int b = blockIdx.z;
  const int split = blockIdx.y;
  const int qbase = blockIdx.x * 32;
  const int lane = threadIdx.x;
  const int half = lane >> 4;
  const int l16 = lane & 15;
  const float* Q = Qg + (size_t)b * NN * DD;
  const float* K = Kg + (size_t)b * NN * DD;
  const float* V = Vg + (size_t)b * NN * DD;

  // B operands (Q^T, 4x16): vgpr0 = dims {0|2} per half, vgpr1 = dims {1|3}.
  v2f Bq0[2], Bq1[2];
#pragma unroll
  for (int s = 0; s < 2; ++s) {
    int qi = qbase + s * 16 + l16;
    if (qi > NN - 1) qi = NN - 1;
    const float4* qp = (const float4*)(Q + (size_t)qi * DD);
    float4 qa = qp[0], qb = qp[1];
    v2f t0 = {half ? qa.z : qa.x, half ? qa.w : qa.y};
    v2f t1 = {half ? qb.z : qb.x, half ? qb.w : qb.y};
    Bq0[s] = t0;
    Bq1[s] = t1;
  }

  v8f O[2];
  O[0] = vzero8(); O[1] = vzero8();
  float m_[2] = {-1e30f, -1e30f};
  float l_[2] = {0.f, 0.f};

  const int cbeg = split * CPS;
  int cend = cbeg + CPS;
  if (cend > NCHUNK) cend = NCHUNK;

  for (int kc = cbeg; kc < cend; ++kc) {
    const int kb = kc * 16;
    int kr = kb + l16;
    if (kr > NN - 1) kr = NN - 1;
    const float4* kp = (const float4*)(K + (size_t)kr * DD);
    float4 ka = kp[0], kbv = kp[1];
    // A operands (K tile, 16x4): single cndmask per component (half is 0/1)
    v2f Ak0 = {half ? ka.z : ka.x, half ? ka.w : ka.y};
    v2f Ak1 = {half ? kbv.z : kbv.x, half ? kbv.w : kbv.y};

    float pp[2][8];
#pragma unroll
    for (int s = 0; s < 2; ++s) {
      v8f S = vzero8();
      S = wmma4(Ak0, Bq0[s], S);   // dims 0..3
      S = wmma4(Ak1, Bq1[s], S);   // dims 4..7 (accumulate)
      float sj[8];
#pragma unroll
      for (int j = 0; j < 8; ++j) sj[j] = S[j] * 0.35355339059327373f;
      if (kb + 16 > NN) {          // remainder-chunk key mask
#pragma unroll
        for (int j = 0; j < 8; ++j)
          if (kb + j + 8 * half >= NN) sj[j] = -1e30f;
      }
      float cm = -1e30f;
#pragma unroll
      for (int j = 0; j < 8; ++j) cm = fmaxf(cm, sj[j]);
      cm = fmaxf(cm, __shfl_xor(cm, 16, 32));      // combine key halves
      const float mn = fmaxf(m_[s], cm);
      const float sc = __expf(m_[s] - mn);
      float rs = 0.f;
#pragma unroll
      for (int j = 0; j < 8; ++j) {
        float pv = __expf(sj[j] - mn);
        pp[s][j] = pv;
        rs += pv;
      }
      rs += __shfl_xor(rs, 16, 32);
      l_[s] = l_[s] * sc + rs;
      m_[s] = mn;
#pragma unroll
      for (int j = 0; j < 8; ++j) O[s][j] *= sc;   // rescale accumulator
    }

    // O^T += V^T(16x4 keys) * P^T(4x16)
#pragma unroll
    for (int c = 0; c < 4; ++c) {
      int vr0 = kb + 4 * c + 2 * half;
      if (vr0 > NN - 1) vr0 = NN - 1;
      int vr1 = vr0 + 1;
      if (vr1 > NN - 1) vr1 = NN - 1;
      float a0 = (l16 < 8) ? V[(size_t)vr0 * DD + l16] : 0.f;
      float a1 = (l16 < 8) ? V[(size_t)vr1 * DD + l16] : 0.f;
      v2f Av = {a0, a1};
      const int base = (c < 2) ? 4 * c : 4 * (c - 2);
#pragma unroll
      for (int s = 0; s < 2; ++s) {
        float t0 = __shfl_xor((c < 2) ? pp[s][base + 2] : pp[s][base + 0], 16, 32);
        float t1 = __shfl_xor((c < 2) ? pp[s][base + 3] : pp[s][base + 1], 16, 32);
        float b0 = (c < 2) ? (half ? t0 : pp[s][base + 0])
                           : (half ? pp[s][base + 2] : t0);
        float b1 = (c < 2) ? (half ? t1 : pp[s][base + 1])
                           : (half ? pp[s][base + 3] : t1);
        v2f Bp = {b0, b1};
        O[s] = wmma4(Av, Bp, O[s]);
      }
    }
  }

  // dims 0..7 for query l16 live in lanes 0..15 across the 8 accumulator VGPRs
  if (half == 0) {
#pragma unroll
    for (int s = 0; s < 2; ++s) {
      int qi = qbase + s * 16 + l16;
      if (qi < NN) {
        float* pb = Part + (((size_t)b * NN + qi) * SPLITS + split) * 12;
#pragma unroll
        for (int j = 0; j < 8; ++j) pb[j] = O[s][j];
        pb[8] = m_[s];
        pb[9] = l_[s];
      }
    }
  }
}

// merge split-K partials: log-sum-exp combine, then normalize
__global__ void attn_merge_kernel(const float* __restrict__ Part,
                                  float* __restrict__ AVg) {
  int idx = blockIdx.x * blockDim.x + threadIdx.x;
  if (idx >= BSZ * NN) return;
  const float* pb = Part + (size_t)idx * SPLITS * 12;
  float M = -1e30f;
#pragma unroll
  for (int sp = 0; sp < SPLITS; ++sp) M = fmaxf(M, pb[sp * 12 + 8]);
  float L = 0.f;
  float O[8];
#pragma unroll
  for (int j = 0; j < 8; ++j) O[j] = 0.f;
#pragma unroll
  for (int sp = 0; sp < SPLITS; ++sp) {
    float w = __expf(pb[sp * 12 + 8] - M);
    L += pb[sp * 12 + 9] * w;
#pragma unroll
    for (int j = 0; j < 8; ++j) O[j] += pb[sp * 12 + j] * w;
  }
  const float inv = 1.f / L;
#pragma unroll
  for (int j = 0; j < 8; ++j) AVg[(size_t)idx * 8 + j] = O[j] * inv;
}

// ---------------------------------------------------------------------------
// Kernel 3: out-proj + residual + LN1 + FFN + residual + LN2 (per node)
// ---------------------------------------------------------------------------
__global__ void enc_post_kernel(const float* __restrict__ AV,
                                const float* __restrict__ V2V,
                                const float* __restrict__ wo, const float* __restrict__ bo,
                                const float* __restrict__ ln1g, const float* __restrict__ ln1b,
                                const float* __restrict__ w1, const float* __restrict__ b1w,
                                const float* __restrict__ w2, const float* __restrict__ b2w,
                                const float* __restrict__ ln2g, const float* __restrict__ ln2b,
                                float* __restrict__ H2) {
  int idx = blockIdx.x * blockDim.x + threadIdx.x;
  if (idx >= BSZ * NN) return;
  const float* WO = wo + LAYER * 64;  const float* BO = bo + LAYER * 8;
  const float* G1 = ln1g + LAYER * 8; const float* Bt1 = ln1b + LAYER * 8;
  const float* W1 = w1 + LAYER * 64;  const float* Bf1 = b1w + LAYER * 8;
  const float* W2 = w2 + LAYER * 64;  const float* Bf2 = b2w + LAYER * 8;
  const float* G2 = ln2g + LAYER * 8; const float* Bt2 = ln2b + LAYER * 8;

  float a[8], x[8];
#pragma unroll
  for (int d = 0; d < 8; ++d) {
    a[d] = AV[(size_t)idx * 8 + d];
    x[d] = V2V[(size_t)idx * 8 + d];
  }
  float r1[8];
#pragma unroll
  for (int d = 0; d < 8; ++d) {
    float o = BO[d];
#pragma unroll
    for (int t = 0; t < 8; ++t) o += a[t] * WO[t * 8 + d];
    r1[d] = x[d] + o;
  }
  float mu = 0.f;
#pragma unroll
  for (int d = 0; d < 8; ++d) mu += r1[d];
  mu *= 0.125f;
  float var = 0.f;
#pragma unroll
  for (int d = 0; d < 8; ++d) { float t = r1[d] - mu; var += t * t; }
  var *= 0.125f;
  float inv = rsqrtf(var + LN_EPS);
  float h[8];
#pragma unroll
  for (int d = 0; d < 8; ++d) h[d] = G1[d] * (r1[d] - mu) * inv + Bt1[d];

  float t1[8];
#pragma unroll
  for (int d = 0; d < 8; ++d) {
    float v = Bf1[d];
#pragma unroll
    for (int t = 0; t < 8; ++t) v += h[t] * W1[t * 8 + d];
    t1[d] = fmaxf(v, 0.f);
  }
  float r2[8];
#pragma unroll
  for (int d = 0; d < 8; ++d) {
    float v = Bf2[d];
#pragma unroll
    for (int t = 0; t < 8; ++t) v += t1[t] * W2[t * 8 + d];
    r2[d] = h[d] + v;
  }
  mu = 0.f;
#pragma unroll
  for (int d = 0; d < 8; ++d) mu += r2[d];
  mu *= 0.125f;
  var = 0.f;
#pragma unroll
  for (int d = 0; d < 8; ++d) { float t = r2[d] - mu; var += t * t; }
  var *= 0.125f;
  inv = rsqrtf(var + LN_EPS);
#pragma unroll
  for (int d = 0; d < 8; ++d)
    H2[(size_t)idx * 8 + d] = G2[d] * (r2[d] - mu) * inv + Bt2[d];
}

// ---------------------------------------------------------------------------
// GCN normalization
// ---------------------------------------------------------------------------
__global__ void deg_init_kernel(float* __restrict__ deg) {
  int n = blockIdx.x * blockDim.x + threadIdx.x;
  if (n < NN) deg[n] = 1.f;  // self-loop
}
__global__ void deg_edge_kernel(const int* __restrict__ dst, float* __restrict__ deg) {
  int e = blockIdx.x * blockDim.x + threadIdx.x;
  if (e < EE) atomicAdd(&deg[dst[e]], 1.f);
}
__global__ void dinv_kernel(float* __restrict__ deg) {
  int n = blockIdx.x * blockDim.x + threadIdx.x;
  if (n < NN) deg[n] = rsqrtf(deg[n]);  // deg >= 1 always
}

// ---------------------------------------------------------------------------
// WMMA node-feature projection: Y[b,n,0:64] = (relu?)X[b,n,0:C] @ W[C,64]
// One wave per 16-node tile; rows kept as float4 so the A-operand build is a
// single cndmask per component (no dynamic register indexing).
// ---------------------------------------------------------------------------
template <int C, bool RELU>
__global__ __launch_bounds__(32) void proj_wmma_kernel(const float* __restrict__ X,
                                                       const float* __restrict__ W,
                                                       float* __restrict__ Y) {
  const int b = blockIdx.y;
  const int nbase = blockIdx.x * 16;
  const int lane = threadIdx.x;
  const int half = lane >> 4;
  const int l16 = lane & 15;
  int row = nbase + l16;
  if (row > NN - 1) row = NN - 1;
  const float4* xr4 = (const float4*)(X + ((size_t)b * NN + row) * C);
  float4 xv4[C / 4];
#pragma unroll
  for (int i = 0; i < C / 4; ++i) {
    float4 t = xr4[i];
    if (RELU) {
      t.x = fmaxf(t.x, 0.f); t.y = fmaxf(t.y, 0.f);
      t.z = fmaxf(t.z, 0.f); t.w = fmaxf(t.w, 0.f);
    }
    xv4[i] = t;
  }
#pragma unroll
  for (int nc = 0; nc < 4; ++nc) {
    v8f acc = vzero8();
#pragma unroll
    for (int kc = 0; kc < C / 4; ++kc) {
      float4 f = xv4[kc];
      v2f A = {half ? f.z : f.x, half ? f.w : f.y};
      const int kr = 4 * kc + 2 * half;
      v2f B = {W[kr * 64 + nc * 16 + l16], W[(kr + 1) * 64 + nc * 16 + l16]};
      acc = wmma4(A, B, acc);
    }
#pragma unroll
    for (int j = 0; j < 8; ++j) {
      int node = nbase + j + 8 * half;
      if (node < NN)
        Y[((size_t)b * NN + node) * 64 + nc * 16 + l16] = acc[j];
    }
  }
}

// g[b,n,h] = bias[h] + dinv[n]^2 * p[b,n,h]   (self-loop + bias init)
__global__ void ginit_kernel(const float* __restrict__ p, const float* __restrict__ dinv,
                             const float* __restrict__ bias, float* __restrict__ g) {
  int idx = blockIdx.x * blockDim.x + threadIdx.x;
  if (idx >= BSZ * NN * HH) return;
  int h = idx % HH;
  int n = (idx / HH) % NN;
  float di = dinv[n];
  g[idx] = bias[h] + di * di * p[idx];
}

// edge scatter: g[b,dst,:] += dinv[src]*dinv[dst] * p[b,src,:]
__global__ void scatter_kernel(const float* __restrict__ p, const int* __restrict__ src,
                               const int* __restrict__ dst, const float* __restrict__ dinv,
                               float* __restrict__ g) {
  int idx = blockIdx.x * blockDim.x + threadIdx.x;
  if (idx >= EE * 8) return;
  int e = idx >> 3;
  int h0 = (idx & 7) * 8;
  int s = src[e], d = dst[e];
  float w = dinv[s] * dinv[d];
#pragma unroll
  for (int b = 0; b < BSZ; ++b) {
    const float* ps = p + ((size_t)b * NN + s) * HH + h0;
    float* gd = g + ((size_t)b * NN + d) * HH + h0;
#pragma unroll
    for (int i = 0; i < 8; ++i) atomicAdd(&gd[i], w * ps[i]);
  }
}

// ---------------------------------------------------------------------------
// fc: out[b,j] = sum_r g2[b,r] * fc_w[r,j] + fc_b[j]  (r = n*64+h, 366336 rows)
// ---------------------------------------------------------------------------
__global__ void out_init_kernel(const float* __restrict__ fcb, float* __restrict__ out) {
  int t = threadIdx.x;
  if (t < 64) out[t] = fcb[t & 31];
}

__global__ __launch_bounds__(256) void fc_kernel(const float* __restrict__ g2,
                                                 const float* __restrict__ fc_w,
                                                 float* __restrict__ out) {
  __shared__ float sacc[64];
  const int t = threadIdx.x;
  for (int i = t; i < 64; i += 256) sacc[i] = 0.f;
  __syncthreads();
  float acc[64];
#pragma unroll
  for (int i = 0; i < 64; ++i) acc[i] = 0.f;
  const int ROWS = NN * HH;
  for (int r = blockIdx.x * 256 + t; r < ROWS; r += gridDim.x * 256) {
    const float x0 = g2[r];
    const float x1 = g2[(size_t)ROWS + r];
    const float* wr = fc_w + (size_t)r * 32;
    __builtin_prefetch(wr + 256 * 32, 0, 1);
#pragma unroll
    for (int j = 0; j < 32; ++j) {
      float w = wr[j];
      acc[j] += x0 * w;
      acc[32 + j] += x1 * w;
    }
  }
#pragma unroll
  for (int j = 0; j < 64; ++j) atomicAdd(&sacc[j], acc[j]);  // ds_add_f32
  __syncthreads();
  if (t < 64) atomicAdd(&out[t], sacc[t]);
}

// ---------------------------------------------------------------------------
extern "C" void kernel_launch(void* const* d_in, const int* in_sizes, int n_in,
                              void* d_out, int out_size, void* d_ws, size_t ws_size,
                              hipStream_t stream) {
  const float* x     = (const float*)d_in[0];
  const int*   eidx  = (const int*)d_in[1];
  const float* lin_w = (const float*)d_in[2];
  const float* lin_b = (const float*)d_in[3];
  const float* wq = (const float*)d_in[4];  const float* bq = (const float*)d_in[5];
  const float* wk = (const float*)d_in[6];  const float* bk = (const float*)d_in[7];
  const float* wv = (const float*)d_in[8];  const float* bv = (const float*)d_in[9];
  const float* wo = (const float*)d_in[10]; const float* bo = (const float*)d_in[11];
  const float* ln1g = (const float*)d_in[12]; const float* ln1b = (const float*)d_in[13];
  const float* w1 = (const float*)d_in[14]; const float* b1 = (const float*)d_in[15];
  const float* w2 = (const float*)d_in[16]; const float* b2 = (const float*)d_in[17];
  const float* ln2g = (const float*)d_in[18]; const float* ln2b = (const float*)d_in[19];
  const float* gcn1_w = (const float*)d_in[20]; const float* gcn1_b = (const float*)d_in[21];
  const float* gcn2_w = (const float*)d_in[22]; const float* gcn2_b = (const float*)d_in[23];
  const float* fc_w = (const float*)d_in[24];   const float* fc_b = (const float*)d_in[25];
  float* out = (float*)d_out;

  const int* src = eidx;
  const int* dst = eidx + EE;

  // workspace layout (floats)
  const size_t NE8 = (size_t)BSZ * NN * 8;
  const size_t NE64 = (size_t)BSZ * NN * 64;
  float* ws = (float*)d_ws;
  float* v2v = ws;
  float* q   = ws + NE8;
  float* k   = ws + 2 * NE8;
  float* v   = ws + 3 * NE8;
  float* av  = ws + 4 * NE8;
  float* h2  = ws + 5 * NE8;
  float* dinv = ws + 6 * NE8;
  float* p1  = ws + 6 * NE8 + 8192;
  float* g1  = p1 + NE64;
  float* p2  = p1 + 2 * NE64;
  float* g2  = p1 + 3 * NE64;
  // split-K attention partials alias p1 (dead until proj1): BSZ*NN*SPLITS*12 floats
  float* avp = p1;

  const int TPB = 256;
  // 1) qkv (layer 3 only)
  qkv_kernel<<<(BSZ * NN + TPB - 1) / TPB, TPB, 0, stream>>>(
      x, lin_w, lin_b, wq, bq, wk, bk, wv, bv, v2v, q, k, v);
  // 2) split-K streaming attention (WMMA f32 16x16x4) + merge
  attn_kernel<<<dim3((NN + 31) / 32, SPLITS, BSZ), 32, 0, stream>>>(q, k, v, avp);
  attn_merge_kernel<<<(BSZ * NN + TPB - 1) / TPB, TPB, 0, stream>>>(avp, av);
  // 3) encoder epilogue
  enc_post_kernel<<<(BSZ * NN + TPB - 1) / TPB, TPB, 0, stream>>>(
      av, v2v, wo, bo, ln1g, ln1b, w1, b1, w2, b2, ln2g, ln2b, h2);
  // 4) GCN normalization
  deg_init_kernel<<<(NN + TPB - 1) / TPB, TPB, 0, stream>>>(dinv);
  deg_edge_kernel<<<(EE + TPB - 1) / TPB, TPB, 0, stream>>>(dst, dinv);
  dinv_kernel<<<(NN + TPB - 1) / TPB, TPB, 0, stream>>>(dinv);
  // 5) gcn1: project (WMMA), init with bias+self-loop, edge scatter
  proj_wmma_kernel<8, false><<<dim3((NN + 15) / 16, BSZ), 32, 0, stream>>>(h2, gcn1_w, p1);
  ginit_kernel<<<((int)(BSZ * NN * HH) + TPB - 1) / TPB, TPB, 0, stream>>>(p1, dinv, gcn1_b, g1);
  scatter_kernel<<<(EE * 8 + TPB - 1) / TPB, TPB, 0, stream>>>(p1, src, dst, dinv, g1);
  // 6) gcn2: relu folded into projection load
  proj_wmma_kernel<64, true><<<dim3((NN + 15) / 16, BSZ), 32, 0, stream>>>(g1, gcn2_w, p2);
  ginit_kernel<<<((int)(BSZ * NN * HH) + TPB - 1) / TPB, TPB, 0, stream>>>(p2, dinv, gcn2_b, g2);
  scatter_kernel<<<(EE * 8 + TPB - 1) / TPB, TPB, 0, stream>>>(p2, src, dst, dinv, g2);
  // 7) fc head
  out_init_kernel<<<1, 64, 0, stream>>>(fc_b, out);
  fc_kernel<<<256, 256, 0, stream>>>(g2, fc_w, out);
}